// MABSink_15599321219682
// MI455X (gfx1250) — compile-verified
//
#include <hip/hip_runtime.h>
#include <hip/hip_bf16.h>

// ---------------- problem constants ----------------
#define Bc   8
#define NQc  1024
#define NKc  1024
#define Dc   512     // DQ = DK = DV
#define Hc   8
#define DSc  64
#define HBc  64      // H * B
#define EPSC     1e-4f
#define INV_EPS  1e4f
#define LOG_MARG (-6.9314718056f)     // -log(1024)
#define SCALEC   0.04419417382f       // 1/sqrt(512) = 1/SQRT_V^2
#define TWO_SCALE 0.08838834765f      // 2/sqrt(512)

typedef float v2f __attribute__((ext_vector_type(2)));
typedef float v8f __attribute__((ext_vector_type(8)));

// D = A(16x4) * B(4x16) + C, fp32, wave32
__device__ __forceinline__ v8f wmma4(v2f a, v2f b, v8f c) {
  return __builtin_amdgcn_wmma_f32_16x16x4_f32(
      /*neg_a=*/false, a, /*neg_b=*/false, b,
      /*c_mod=*/(short)0, c, /*reuse_a=*/false, /*reuse_b=*/false);
}

// ---------------------------------------------------------------------------
// Kernel 1: out[m,n] = X[m,:] . W[n,:] + b[n], written head-split:
//           outH[(h*B + bb)][nn][ds]  with h = n/64, ds = n%64, m = bb*NQ+nn
// One wave per 16x16 tile; K = 512 -> 128 WMMA f32 16x16x4 ops per tile.
// ---------------------------------------------------------------------------
__global__ __launch_bounds__(32)
void proj_headsplit(const float* __restrict__ X, const float* __restrict__ W,
                    const float* __restrict__ bias, float* __restrict__ outH,
                    int nTilesN) {
  const int lane = threadIdx.x;
  const int lo = lane & 15, hi = lane >> 4;
  const int mt = blockIdx.x / nTilesN;
  const int nt = blockIdx.x % nTilesN;
  const int m0 = mt * 16, n0 = nt * 16;

  const float* Ar = X + (size_t)(m0 + lo) * Dc;   // A row for this lane
  const float* Br = W + (size_t)(n0 + lo) * Dc;   // W row (B column) for this lane
  const float bval = bias[n0 + lo];

  v8f acc;
#pragma unroll
  for (int r = 0; r < 8; ++r) acc[r] = bval;

  for (int k = 0; k < Dc; k += 4) {
    v2f a = *(const v2f*)(Ar + k + hi * 2);   // A[M=lo, K = k + hi*2 + {0,1}]
    v2f b = *(const v2f*)(Br + k + hi * 2);   // B[K = k + hi*2 + {0,1}, N=lo]
    acc = wmma4(a, b, acc);
  }

#pragma unroll
  for (int r = 0; r < 8; ++r) {
    const int row = m0 + r + hi * 8;        // flat (bb,nn)
    const int col = n0 + lo;
    const int bb = row >> 10, nn = row & 1023;
    const int h = col >> 6, ds = col & 63;
    outH[(size_t)((h * Bc + bb) * NQc + nn) * DSc + ds] = acc[r];
  }
}

// ---------------------------------------------------------------------------
// Kernel 2: scaled squared row norms of qh and kh:  n2 = ||x||^2 / sqrt(512)
// ---------------------------------------------------------------------------
__global__ __launch_bounds__(256)
void row_norms(const float* __restrict__ qh, const float* __restrict__ kh,
               float* __restrict__ q2, float* __restrict__ k2) {
  const int idx = blockIdx.x * 256 + threadIdx.x;   // 0 .. 2*65536
  const int sel = idx >> 16;
  const int row = idx & 65535;
  const float* src = (sel ? kh : qh) + (size_t)row * DSc;
  float s = 0.f;
#pragma unroll
  for (int i = 0; i < 16; ++i) {
    float4 t = ((const float4*)src)[i];
    s += t.x * t.x + t.y * t.y + t.z * t.z + t.w * t.w;
  }
  (sel ? k2 : q2)[row] = s * SCALEC;
}

// ---------------------------------------------------------------------------
// Kernel 3: fused Sinkhorn (3 iters, C recomputed via WMMA each sweep) +
//           apply O = q + P @ v, one workgroup (8 waves) per head-batch m.
// LDS: K_m (raw, 256KB, staged via async global->LDS DMA) + norms + u,v +
//      per-wave P scratch = ~280KB of the 320KB WGP LDS.
// ---------------------------------------------------------------------------
__global__ __launch_bounds__(256)
void sinkhorn_apply(const float* __restrict__ qh, const float* __restrict__ kh,
                    const float* __restrict__ vh, const float* __restrict__ q2g,
                    const float* __restrict__ k2g, float* __restrict__ Obuf) {
  extern __shared__ float lds[];
  float* kS  = lds;              // [1024][64] raw K_m
  float* k2s = kS + NKc * DSc;   // [1024]
  float* q2s = k2s + NKc;        // [1024]
  float* uS  = q2s + NQc;        // [1024]
  float* vS  = uS + NQc;         // [1024]
  float* Pl  = vS + NKc;         // 8 waves * 256 : P-tile transpose scratch

  const int m = blockIdx.x;
  const float* qm = qh + (size_t)m * NQc * DSc;
  const float* km = kh + (size_t)m * NKc * DSc;
  const float* vm = vh + (size_t)m * NKc * DSc;
  const int tid = threadIdx.x;

  // ---- stage K_m into LDS with CDNA5 async global->LDS copies (no VGPR
  //      round-trip); 16384 b128 transfers, tracked by ASYNCcnt ----
  for (int it = 0; it < (NKc * DSc) / (256 * 4); ++it) {   // 64 iterations
    const int idx = (it * 256 + tid) * 4;                  // float index, 16B aligned
    unsigned ldsaddr = (unsigned)(uintptr_t)(kS + idx);    // LDS byte offset
    const float* gaddr = km + idx;
    asm volatile("global_load_async_to_lds_b128 %0, %1, off"
                 :: "v"(ldsaddr), "v"(gaddr) : "memory");
  }
  for (int i = tid; i < NKc; i += 256) {
    k2s[i] = k2g[m * NKc + i];
    q2s[i] = q2g[m * NQc + i];
    uS[i] = 0.f;
    vS[i] = 0.f;
  }
  asm volatile("s_wait_asynccnt 0x0" ::: "memory");
  __syncthreads();

  const int wave = tid >> 5;
  const int lane = tid & 31;
  const int lo = lane & 15, hi = lane >> 4;
  float* Pw = Pl + wave * 256;

  for (int it = 0; it < 3; ++it) {
    // ---- u_i = eps*(log_a - LSE_j((v_j - C_ij)/eps)) ----
    for (int t = wave; t < NQc / 16; t += 8) {
      const int i0 = t * 16;
      float rmax[8], rsum[8];
#pragma unroll
      for (int r = 0; r < 8; ++r) { rmax[r] = -3.0e38f; rsum[r] = 0.f; }
      for (int jt = 0; jt < NKc / 16; ++jt) {
        const int j0 = jt * 16;
        v8f acc = {};
#pragma unroll
        for (int kk = 0; kk < DSc; kk += 4) {
          v2f a = *(const v2f*)(qm + (i0 + lo) * DSc + kk + hi * 2);
          v2f b = *(const v2f*)(kS + (j0 + lo) * DSc + kk + hi * 2);
          acc = wmma4(a, b, acc);            // raw q.k
        }
        const int jj = j0 + lo;
        const float k2v = k2s[jj];
        const float vv = vS[jj];
#pragma unroll
        for (int r = 0; r < 8; ++r) {
          const int ii = i0 + r + hi * 8;
          float c = sqrtf(fmaxf(q2s[ii] + k2v - TWO_SCALE * acc[r], 0.f));
          float z = (vv - c) * INV_EPS;
          float mo = rmax[r];
          if (z > mo) { rsum[r] = rsum[r] * __expf(mo - z) + 1.f; rmax[r] = z; }
          else        { rsum[r] += __expf(z - mo); }
        }
      }
#pragma unroll
      for (int r = 0; r < 8; ++r) {
        float Mx = rmax[r], Sx = rsum[r];
#pragma unroll
        for (int off = 1; off < 16; off <<= 1) {   // butterfly within 16-lane half
          float Mo = __shfl_xor(Mx, off, 32);
          float So = __shfl_xor(Sx, off, 32);
          float nm = fmaxf(Mx, Mo);
          Sx = Sx * __expf(Mx - nm) + So * __expf(Mo - nm);
          Mx = nm;
        }
        if (lo == 0) {
          const int ii = i0 + r + hi * 8;
          uS[ii] = EPSC * (LOG_MARG - (Mx + __logf(Sx)));
        }
      }
    }
    __syncthreads();
    // ---- v_j = eps*(log_b - LSE_i((u_i - C_ij)/eps)) : roles swapped ----
    for (int t = wave; t < NKc / 16; t += 8) {
      const int j0 = t * 16;
      float rmax[8], rsum[8];
#pragma unroll
      for (int r = 0; r < 8; ++r) { rmax[r] = -3.0e38f; rsum[r] = 0.f; }
      for (int ct = 0; ct < NQc / 16; ++ct) {
        const int c0 = ct * 16;
        v8f acc = {};
#pragma unroll
        for (int kk = 0; kk < DSc; kk += 4) {
          v2f a = *(const v2f*)(kS + (j0 + lo) * DSc + kk + hi * 2);
          v2f b = *(const v2f*)(qm + (c0 + lo) * DSc + kk + hi * 2);
          acc = wmma4(a, b, acc);            // rows = j, cols = i
        }
        const int ii = c0 + lo;
        const float q2v = q2s[ii];
        const float uu = uS[ii];
#pragma unroll
        for (int r = 0; r < 8; ++r) {
          const int jj = j0 + r + hi * 8;
          float c = sqrtf(fmaxf(k2s[jj] + q2v - TWO_SCALE * acc[r], 0.f));
          float z = (uu - c) * INV_EPS;
          float mo = rmax[r];
          if (z > mo) { rsum[r] = rsum[r] * __expf(mo - z) + 1.f; rmax[r] = z; }
          else        { rsum[r] += __expf(z - mo); }
        }
      }
#pragma unroll
      for (int r = 0; r < 8; ++r) {
        float Mx = rmax[r], Sx = rsum[r];
#pragma unroll
        for (int off = 1; off < 16; off <<= 1) {
          float Mo = __shfl_xor(Mx, off, 32);
          float So = __shfl_xor(Sx, off, 32);
          float nm = fmaxf(Mx, Mo);
          Sx = Sx * __expf(Mx - nm) + So * __expf(Mo - nm);
          Mx = nm;
        }
        if (lo == 0) {
          const int jj = j0 + r + hi * 8;
          vS[jj] = EPSC * (LOG_MARG - (Mx + __logf(Sx)));
        }
      }
    }
    __syncthreads();
  }

  // ---- O_i = q_i + sum_j exp((u_i + v_j - C_ij)/eps) * v_j  (P@V via WMMA) ----
  const int h = m / Bc, bb = m % Bc;    // m = h*B + b
  for (int t = wave; t < NQc / 16; t += 8) {
    const int i0 = t * 16;
    v8f oacc[4];
#pragma unroll
    for (int dt = 0; dt < 4; ++dt) oacc[dt] = {};
    for (int jt = 0; jt < NKc / 16; ++jt) {
      const int j0 = jt * 16;
      v8f acc = {};
#pragma unroll
      for (int kk = 0; kk < DSc; kk += 4) {
        v2f a = *(const v2f*)(qm + (i0 + lo) * DSc + kk + hi * 2);
        v2f b = *(const v2f*)(kS + (j0 + lo) * DSc + kk + hi * 2);
        acc = wmma4(a, b, acc);
      }
      const int jj = j0 + lo;
      const float k2v = k2s[jj];
      const float vv = vS[jj];
#pragma unroll
      for (int r = 0; r < 8; ++r) {                // P tile -> LDS (row-major 16x16)
        const int ii = i0 + r + hi * 8;
        float c = sqrtf(fmaxf(q2s[ii] + k2v - TWO_SCALE * acc[r], 0.f));
        Pw[(r + hi * 8) * 16 + lo] = __expf((uS[ii] + vv - c) * INV_EPS);
      }
      __syncthreads();   // uniform trip counts: all waves do 8 * 64 iterations
#pragma unroll
      for (int kk = 0; kk < 16; kk += 4) {         // K = 16 (j) -> 4 wmma steps
        v2f aP = *(const v2f*)(Pw + lo * 16 + kk + hi * 2);  // A-layout reload of P
        const int kr = j0 + kk + hi * 2;
#pragma unroll
        for (int dt = 0; dt < 4; ++dt) {           // DS = 64 -> 4 N-tiles
          v2f bV;
          bV.x = vm[(kr + 0) * DSc + dt * 16 + lo];
          bV.y = vm[(kr + 1) * DSc + dt * 16 + lo];
          oacc[dt] = wmma4(aP, bV, oacc[dt]);
        }
      }
      __syncthreads();
    }
#pragma unroll
    for (int dt = 0; dt < 4; ++dt) {
#pragma unroll
      for (int r = 0; r < 8; ++r) {
        const int ii = i0 + r + hi * 8;
        const int d = dt * 16 + lo;
        const float val = qm[ii * DSc + d] + oacc[dt][r];
        Obuf[((size_t)bb * NQc + ii) * Dc + h * DSc + d] = val;   // merged heads
      }
    }
  }
}

// ---------------------------------------------------------------------------
// Kernel 4: out = O + relu(O @ Wo^T + bo)
// ---------------------------------------------------------------------------
__global__ __launch_bounds__(32)
void out_proj(const float* __restrict__ O, const float* __restrict__ W,
              const float* __restrict__ bias, float* __restrict__ out,
              int nTilesN) {
  const int lane = threadIdx.x;
  const int lo = lane & 15, hi = lane >> 4;
  const int mt = blockIdx.x / nTilesN;
  const int nt = blockIdx.x % nTilesN;
  const int m0 = mt * 16, n0 = nt * 16;

  const float* Ar = O + (size_t)(m0 + lo) * Dc;
  const float* Br = W + (size_t)(n0 + lo) * Dc;
  const float bval = bias[n0 + lo];

  v8f acc;
#pragma unroll
  for (int r = 0; r < 8; ++r) acc[r] = bval;

  for (int k = 0; k < Dc; k += 4) {
    v2f a = *(const v2f*)(Ar + k + hi * 2);
    v2f b = *(const v2f*)(Br + k + hi * 2);
    acc = wmma4(a, b, acc);
  }

#pragma unroll
  for (int r = 0; r < 8; ++r) {
    const int row = m0 + r + hi * 8;
    const int col = n0 + lo;
    const float o = O[(size_t)row * Dc + col];
    out[(size_t)row * Dc + col] = o + fmaxf(acc[r], 0.f);
  }
}

// ---------------------------------------------------------------------------
extern "C" void kernel_launch(void* const* d_in, const int* in_sizes, int n_in,
                              void* d_out, int out_size, void* d_ws, size_t ws_size,
                              hipStream_t stream) {
  (void)in_sizes; (void)n_in; (void)out_size; (void)ws_size;
  const float* Q  = (const float*)d_in[0];
  const float* K  = (const float*)d_in[1];
  const float* Wq = (const float*)d_in[2];
  const float* bq = (const float*)d_in[3];
  const float* Wk = (const float*)d_in[4];
  const float* bk = (const float*)d_in[5];
  const float* Wv = (const float*)d_in[6];
  const float* bv = (const float*)d_in[7];
  const float* Wo = (const float*)d_in[8];
  const float* bo = (const float*)d_in[9];
  float* out = (float*)d_out;

  float* ws = (float*)d_ws;
  float* qh = ws;                                   // [64][1024][64]
  float* kh = qh + (size_t)HBc * NQc * DSc;         // [64][1024][64]
  float* vh = kh + (size_t)HBc * NKc * DSc;         // [64][1024][64]
  float* Ob = vh + (size_t)HBc * NKc * DSc;         // [8][1024][512]
  float* q2 = Ob + (size_t)Bc * NQc * Dc;           // [64*1024]
  float* k2 = q2 + (size_t)HBc * NQc;               // [64*1024]

  const int M = Bc * NQc;          // 8192
  const int nTilesN = Dc / 16;     // 32
  dim3 gProj((M / 16) * nTilesN);  // 16384 tiles, one wave each

  proj_headsplit<<<gProj, 32, 0, stream>>>(Q, Wq, bq, qh, nTilesN);
  proj_headsplit<<<gProj, 32, 0, stream>>>(K, Wk, bk, kh, nTilesN);
  proj_headsplit<<<gProj, 32, 0, stream>>>(K, Wv, bv, vh, nTilesN);

  row_norms<<<(2 * HBc * NQc) / 256, 256, 0, stream>>>(qh, kh, q2, k2);

  const size_t ldsBytes =
      (size_t)(NKc * DSc + NKc + NQc + NQc + NKc + 8 * 256) * sizeof(float);
  sinkhorn_apply<<<HBc, 256, ldsBytes, stream>>>(qh, kh, vh, q2, k2, Ob);

  out_proj<<<gProj, 32, 0, stream>>>(Ob, Wo, bo, out, nTilesN);
}